// Con_Deep_19250043420783
// MI455X (gfx1250) — compile-verified
//
#include <hip/hip_runtime.h>
#include <math.h>

// Problem constants (from reference)
#define NCLS   19
#define CDIM   512
#define HW     16384          // 128*128
#define BATCH  8
#define NPIX   (BATCH*HW)     // 131072
#define TEMP_INV 10.0f

// Reduction kernel tiling
#define TPB    256
#define PXT    4                          // pixels per thread (float4 loads)
#define PIX_PER_BLOCK (TPB*PXT)           // 1024
#define NTILES (NPIX/PIX_PER_BLOCK)       // 128
#define CCHUNK 256
#define NCH    (CDIM/CCHUNK)              // 2
#define ASTRIDE (CCHUNK+1)                // 257: bank = (label + c) % 64

typedef __attribute__((ext_vector_type(4))) float v4f;
typedef __attribute__((ext_vector_type(4))) int   v4i;
typedef __attribute__((ext_vector_type(2))) float v2f;
typedef __attribute__((ext_vector_type(8))) float v8f;

// Workspace layout (floats): [0,19*512) sum_s | [19*512,2*19*512) sum_t | +19 counts
#define WS_FLOATS (2*NCLS*CDIM + NCLS)

__global__ void k_zero_ws(float* ws, int n) {
    int i = blockIdx.x * blockDim.x + threadIdx.x;
    if (i < n) ws[i] = 0.0f;
}

// Streaming segment-sum: 512 MB read once -> memory-bound, target full HBM BW.
__global__ __launch_bounds__(TPB) void k_reduce(const float* __restrict__ fs,
                                                const float* __restrict__ ft,
                                                const int*   __restrict__ lab,
                                                float* __restrict__ ws) {
    __shared__ float acc[2][NCLS][ASTRIDE];
    __shared__ float lcnt[NCLS];
    const int tid = threadIdx.x;

    for (int i = tid; i < 2 * NCLS * ASTRIDE; i += TPB) ((float*)acc)[i] = 0.0f;
    if (tid < NCLS) lcnt[tid] = 0.0f;
    __syncthreads();

    const int bid  = blockIdx.x;
    const int tile = bid >> 1;   // 0..127 pixel tile
    const int cc   = bid & 1;    // channel half

    const int p0 = tile * PIX_PER_BLOCK + tid * PXT;   // 16B aligned
    const int b  = p0 / HW;
    const int hw = p0 % HW;

    const v4i lv = *(const v4i*)(lab + p0);
    const int l0 = lv.x, l1 = lv.y, l2 = lv.z, l3 = lv.w;

    if (cc == 0) {
        atomicAdd(&lcnt[l0], 1.0f); atomicAdd(&lcnt[l1], 1.0f);
        atomicAdd(&lcnt[l2], 1.0f); atomicAdd(&lcnt[l3], 1.0f);
    }

    const long base = (long)b * CDIM * HW + (long)cc * CCHUNK * HW + hw;
    const float* ps = fs + base;
    const float* pt = ft + base;

    for (int c = 0; c < CCHUNK; ++c) {
        // read-once data: non-temporal 128-bit loads, don't pollute L2
        v4f vs = __builtin_nontemporal_load((const v4f*)(ps + (long)c * HW));
        v4f vt = __builtin_nontemporal_load((const v4f*)(pt + (long)c * HW));
        atomicAdd(&acc[0][l0][c], vs.x);
        atomicAdd(&acc[0][l1][c], vs.y);
        atomicAdd(&acc[0][l2][c], vs.z);
        atomicAdd(&acc[0][l3][c], vs.w);
        atomicAdd(&acc[1][l0][c], vt.x);
        atomicAdd(&acc[1][l1][c], vt.y);
        atomicAdd(&acc[1][l2][c], vt.z);
        atomicAdd(&acc[1][l3][c], vt.w);
    }
    __syncthreads();

    float* gs = ws;
    float* gt = ws + NCLS * CDIM;
    float* gc = ws + 2 * NCLS * CDIM;
    for (int i = tid; i < 2 * NCLS * CCHUNK; i += TPB) {
        const int s = i / (NCLS * CCHUNK);
        const int r = i - s * (NCLS * CCHUNK);
        const int k = r / CCHUNK;
        const int c = r - k * CCHUNK;
        float* g = s ? gt : gs;
        unsafeAtomicAdd(&g[k * CDIM + cc * CCHUNK + c], acc[s][k][c]);
    }
    if (cc == 0 && tid < NCLS) unsafeAtomicAdd(&gc[tid], lcnt[tid]);
}

// Finalize: means, L2-normalize, 19x19 Gram via V_WMMA_F32_16X16X4_F32, loss.
#define MSTRIDE 513   // bank = (row + c) % 64 -> conflict-free per-row walks

__global__ __launch_bounds__(256) void k_final(const float* __restrict__ ws,
                                               float* __restrict__ out) {
    __shared__ float mS[NCLS * MSTRIDE];
    __shared__ float mT[NCLS * MSTRIDE];
    __shared__ float cnt[NCLS];
    __shared__ float inv[2][NCLS];
    __shared__ float logits[32 * 33];
    const int tid = threadIdx.x;

    if (tid < NCLS) cnt[tid] = ws[2 * NCLS * CDIM + tid];
    for (int i = tid; i < 32 * 33; i += 256) logits[i] = 0.0f;
    __syncthreads();

    for (int i = tid; i < NCLS * CDIM; i += 256) {
        const int k = i / CDIM, c = i - k * CDIM;
        const float d = fmaxf(cnt[k], 1.0f);
        mS[k * MSTRIDE + c] = ws[i] / d;
        mT[k * MSTRIDE + c] = ws[NCLS * CDIM + i] / d;
    }
    __syncthreads();

    if (tid < NCLS) {
        float ss = 0.0f, st = 0.0f;
        for (int c = 0; c < CDIM; ++c) {
            const float a = mS[tid * MSTRIDE + c]; ss += a * a;
            const float b = mT[tid * MSTRIDE + c]; st += b * b;
        }
        inv[0][tid] = 1.0f / fmaxf(sqrtf(ss), 1e-12f);
        inv[1][tid] = 1.0f / fmaxf(sqrtf(st), 1e-12f);
    }
    __syncthreads();

    for (int i = tid; i < NCLS * CDIM; i += 256) {
        const int k = i / CDIM, c = i - k * CDIM;
        mS[k * MSTRIDE + c] *= inv[0][k];
        mT[k * MSTRIDE + c] *= inv[1][k];
    }
    __syncthreads();

    if (tid < 32) {  // single wave32 does the padded 32x32x512 Gram matrix
        const int lane = tid;
        const int half = lane >> 4;     // K sub-phase per ISA A/B layout
        const int lm   = lane & 15;
#if __has_builtin(__builtin_amdgcn_wmma_f32_16x16x4_f32)
        for (int mt = 0; mt < 2; ++mt) {
            for (int nt = 0; nt < 2; ++nt) {
                v8f accv = {0.f,0.f,0.f,0.f,0.f,0.f,0.f,0.f};
                const int M = mt * 16 + lm;   // A row for this lane
                const int N = nt * 16 + lm;   // B col for this lane
                for (int k0 = 0; k0 < CDIM; k0 += 4) {
                    const int ka = k0 + 2 * half;
                    v2f a, b;
                    a.x = (M < NCLS) ? mS[M * MSTRIDE + ka]     : 0.0f;
                    a.y = (M < NCLS) ? mS[M * MSTRIDE + ka + 1] : 0.0f;
                    b.x = (N < NCLS) ? mT[N * MSTRIDE + ka]     : 0.0f;
                    b.y = (N < NCLS) ? mT[N * MSTRIDE + ka + 1] : 0.0f;
                    accv = __builtin_amdgcn_wmma_f32_16x16x4_f32(
                        false, a, false, b, (short)0, accv, false, false);
                }
                for (int v = 0; v < 8; ++v) {
                    const int row = mt * 16 + v + 8 * half;   // C/D VGPR layout
                    const int col = nt * 16 + lm;
                    logits[row * 33 + col] = accv[v];
                }
            }
        }
#else
        for (int idx = lane; idx < NCLS * NCLS; idx += 32) {
            const int r = idx / NCLS, c = idx - r * NCLS;
            float s = 0.0f;
            for (int k = 0; k < CDIM; ++k) s += mS[r * MSTRIDE + k] * mT[c * MSTRIDE + k];
            logits[r * 33 + c] = s;
        }
#endif
    }
    __syncthreads();

    if (tid == 0) {
        float lsum = 0.0f; int np = 0;
        for (int k = 0; k < NCLS; ++k) {
            if (cnt[k] <= 0.0f) continue;
            ++np;
            float mx = -1e30f;
            for (int n = 0; n < NCLS; ++n) {
                const float v = logits[k * 33 + n] * TEMP_INV;
                mx = fmaxf(mx, v);
            }
            float se = 0.0f;
            for (int n = 0; n < NCLS; ++n)
                se += __expf(logits[k * 33 + n] * TEMP_INV - mx);
            const float lse = mx + __logf(se);
            lsum += logits[k * 33 + k] * TEMP_INV - lse;
        }
        out[0] = -(lsum / (float)np);
    }
}

extern "C" void kernel_launch(void* const* d_in, const int* in_sizes, int n_in,
                              void* d_out, int out_size, void* d_ws, size_t ws_size,
                              hipStream_t stream) {
    const float* fs  = (const float*)d_in[0];
    const float* ft  = (const float*)d_in[1];
    const int*   lab = (const int*)d_in[2];
    float* out = (float*)d_out;
    float* ws  = (float*)d_ws;

    k_zero_ws<<<(WS_FLOATS + 255) / 256, 256, 0, stream>>>(ws, WS_FLOATS);
    k_reduce<<<NTILES * NCH, TPB, 0, stream>>>(fs, ft, lab, ws);
    k_final<<<1, 256, 0, stream>>>(ws, out);
}